// DictionaryLearningOMP_76914274337306
// MI455X (gfx1250) — compile-verified
//
#include <hip/hip_runtime.h>
#include <hip/hip_bf16.h>

typedef __attribute__((ext_vector_type(2))) float v2f;
typedef __attribute__((ext_vector_type(8))) float v8f;

#define DIMX 256
#define KATOMS 512
#define NCOLS 65536

// ---------------------------------------------------------------------------
// Kernel 1: Gram matrix G[i][j] = sum_k dict[k][i] * dict[k][j]   (256x256)
// ---------------------------------------------------------------------------
__global__ __launch_bounds__(256) void gram256_kernel(
    const float* __restrict__ dict,  // [KATOMS][DIMX]
    float* __restrict__ G)           // [DIMX][DIMX]
{
    const int i = blockIdx.x;    // 0..255
    const int j = threadIdx.x;   // 0..255
    float s = 0.0f;
    for (int k = 0; k < KATOMS; ++k) {
        s += dict[(size_t)k * DIMX + i] * dict[(size_t)k * DIMX + j];
    }
    G[(size_t)i * DIMX + j] = s;
}

// ---------------------------------------------------------------------------
// Kernel 2: in-place Gauss-Jordan inversion of SPD 256x256 in LDS
// (256 KB matrix + 1 KB pivot column; dynamic LDS, one workgroup)
// ---------------------------------------------------------------------------
extern __shared__ float inv_smem[];
__global__ __launch_bounds__(256) void invert256_kernel(float* __restrict__ G)
{
    float* A    = inv_smem;                 // [256][256]
    float* fcol = inv_smem + DIMX * DIMX;   // [256]
    const int j = threadIdx.x;              // this thread owns column j

    for (int i = 0; i < DIMX; ++i)
        A[(size_t)i * DIMX + j] = G[(size_t)i * DIMX + j];
    __syncthreads();

    for (int p = 0; p < DIMX; ++p) {
        // snapshot pivot column (cross-thread read, protected by barriers)
        fcol[j] = A[(size_t)j * DIMX + p];
        __syncthreads();

        const float pivinv = 1.0f / fcol[p];
        // scale pivot row (thread j only touches its own column j)
        const float sr = (j == p) ? pivinv : A[(size_t)p * DIMX + j] * pivinv;
        A[(size_t)p * DIMX + j] = sr;

        // eliminate all other rows; each (i,j) touched only by thread j,
        // f comes from the read-only snapshot -> no barrier needed in-loop
        for (int i = 0; i < DIMX; ++i) {
            if (i == p) continue;
            const float f   = fcol[i];
            const float aij = (j == p) ? 0.0f : A[(size_t)i * DIMX + j];
            A[(size_t)i * DIMX + j] = aij - f * sr;
        }
        __syncthreads();
    }

    for (int i = 0; i < DIMX; ++i)
        G[(size_t)i * DIMX + j] = A[(size_t)i * DIMX + j];
}

// ---------------------------------------------------------------------------
// Kernel 3: Dpinv[k][d] = sum_j dict[k][j] * Ginv[j][d]   (512x256)
// ---------------------------------------------------------------------------
__global__ __launch_bounds__(256) void dpinv_kernel(
    const float* __restrict__ dict,   // [KATOMS][DIMX]
    const float* __restrict__ Ginv,   // [DIMX][DIMX]
    float* __restrict__ Dpinv)        // [KATOMS][DIMX]
{
    const int k = blockIdx.x;    // 0..511
    const int d = threadIdx.x;   // 0..255
    float s = 0.0f;
    for (int jj = 0; jj < DIMX; ++jj) {
        s += dict[(size_t)k * DIMX + jj] * Ginv[(size_t)jj * DIMX + d];
    }
    Dpinv[(size_t)k * DIMX + d] = s;
}

// ---------------------------------------------------------------------------
// Kernel 4: out[n][k] = sum_d Dpinv[k][d] * z_e[d][n] via V_WMMA_F32_16X16X4_F32
// Each wave: one 32(n) x 64(k) super-tile -> 8 accumulators.
// Per K-step (d += 4): 2 A fragments (4 x b32) + 4 B fragments (4 x b64)
// feed 8 WMMAs => 1:1 WMMA:VMEM issue, each B reused 2x, each A reused 4x.
// ---------------------------------------------------------------------------
__global__ __launch_bounds__(256) void omp_gemm_wmma_kernel(
    const float* __restrict__ z_e,    // [DIMX][NCOLS]
    const float* __restrict__ dpinv,  // [KATOMS][DIMX]
    float* __restrict__ out)          // [NCOLS][KATOMS]
{
    const int lane = threadIdx.x & 31;
    const int wave = threadIdx.x >> 5;
    const int flat = blockIdx.x * (blockDim.x >> 5) + wave;

    const int KT    = KATOMS / 64;        // 8 k-super-tiles
    const int ktile = flat % KT;
    const int ntile = flat / KT;          // 0..2047
    const int n0 = ntile * 32;
    const int k0 = ktile * 64;

    const int h = lane >> 4;              // lane half (0/1)
    const int l = lane & 15;              // lane within half

    v8f acc00 = {}, acc01 = {}, acc02 = {}, acc03 = {};
    v8f acc10 = {}, acc11 = {}, acc12 = {}, acc13 = {};

    // A: a.v[r] = z_e[(d0 + 2h + r)*NCOLS + n0 (+16 for frag 1) + l]
    const float* aptr = z_e + (size_t)(2 * h) * NCOLS + (n0 + l);
    // B: b.v[r] = dpinv[(k + l)*DIMX + d0 + 2h + r] -> contiguous float2
    const float* bbase = dpinv + (size_t)(k0 + l) * DIMX + 2 * h;

    #pragma unroll 2
    for (int d0 = 0; d0 < DIMX; d0 += 4) {
        v2f a0, a1;
        a0.x = aptr[0];
        a0.y = aptr[NCOLS];
        a1.x = aptr[16];
        a1.y = aptr[NCOLS + 16];
        aptr += (size_t)4 * NCOLS;

        v2f b0 = *(const v2f*)(bbase + d0);
        v2f b1 = *(const v2f*)(bbase + 16 * DIMX + d0);
        v2f b2 = *(const v2f*)(bbase + 32 * DIMX + d0);
        v2f b3 = *(const v2f*)(bbase + 48 * DIMX + d0);

        acc00 = __builtin_amdgcn_wmma_f32_16x16x4_f32(false, a0, false, b0,
                                                      (short)0, acc00, false, false);
        acc01 = __builtin_amdgcn_wmma_f32_16x16x4_f32(false, a0, false, b1,
                                                      (short)0, acc01, false, false);
        acc02 = __builtin_amdgcn_wmma_f32_16x16x4_f32(false, a0, false, b2,
                                                      (short)0, acc02, false, false);
        acc03 = __builtin_amdgcn_wmma_f32_16x16x4_f32(false, a0, false, b3,
                                                      (short)0, acc03, false, false);
        acc10 = __builtin_amdgcn_wmma_f32_16x16x4_f32(false, a1, false, b0,
                                                      (short)0, acc10, false, false);
        acc11 = __builtin_amdgcn_wmma_f32_16x16x4_f32(false, a1, false, b1,
                                                      (short)0, acc11, false, false);
        acc12 = __builtin_amdgcn_wmma_f32_16x16x4_f32(false, a1, false, b2,
                                                      (short)0, acc12, false, false);
        acc13 = __builtin_amdgcn_wmma_f32_16x16x4_f32(false, a1, false, b3,
                                                      (short)0, acc13, false, false);
    }

    // C/D layout: VGPR v holds row m = v + 8h, column l
    float* obase0 = out + (size_t)(n0 + 8 * h) * KATOMS + (k0 + l);
    float* obase1 = obase0 + (size_t)16 * KATOMS;
    #pragma unroll
    for (int v = 0; v < 8; ++v) {
        obase0[(size_t)v * KATOMS + 0]  = acc00[v];
        obase0[(size_t)v * KATOMS + 16] = acc01[v];
        obase0[(size_t)v * KATOMS + 32] = acc02[v];
        obase0[(size_t)v * KATOMS + 48] = acc03[v];
        obase1[(size_t)v * KATOMS + 0]  = acc10[v];
        obase1[(size_t)v * KATOMS + 16] = acc11[v];
        obase1[(size_t)v * KATOMS + 32] = acc12[v];
        obase1[(size_t)v * KATOMS + 48] = acc13[v];
    }
}

// ---------------------------------------------------------------------------
extern "C" void kernel_launch(void* const* d_in, const int* in_sizes, int n_in,
                              void* d_out, int out_size, void* d_ws, size_t ws_size,
                              hipStream_t stream) {
    const float* z_e  = (const float*)d_in[0];   // [256][65536]
    const float* dict = (const float*)d_in[1];   // [512][256]
    float* out = (float*)d_out;                  // [65536][512]

    float* G     = (float*)d_ws;                 // 256*256 floats
    float* Dpinv = G + DIMX * DIMX;              // 512*256 floats

    // 1) Gram matrix
    gram256_kernel<<<DIMX, 256, 0, stream>>>(dict, G);

    // 2) Invert in LDS (256KB matrix + 1KB pivot column dynamic LDS)
    invert256_kernel<<<1, 256, (DIMX * DIMX + DIMX) * sizeof(float), stream>>>(G);

    // 3) Dpinv = dict @ Ginv
    dpinv_kernel<<<KATOMS, 256, 0, stream>>>(dict, G, Dpinv);

    // 4) Main GEMM: 2048 n-tiles x 8 k-supertiles = 16384 waves / 8 per block
    const int total_waves = (NCOLS / 32) * (KATOMS / 64);
    omp_gemm_wmma_kernel<<<total_waves / 8, 256, 0, stream>>>(z_e, Dpinv, out);
}